// VAE_EP_fullcov_55757265437139
// MI455X (gfx1250) — compile-verified
//
#include <hip/hip_runtime.h>
#include <hip/hip_bf16.h>
#include <math.h>

#define EPS 1e-6f
#define INV_SQRT_2PI 0.3989422804014327f

typedef __attribute__((ext_vector_type(2))) float v2f;
typedef __attribute__((ext_vector_type(8))) float v8f;

__constant__ float GAMMA_SIGMOID[10] = {
    0.519483084417772f, 0.357944855434941f, 0.723301195883257f, 0.474918009444542f,
    0.519483382721337f, 0.357945091498072f, 0.519481351904163f, 0.357945544542554f,
    0.242049896394596f, 0.357944917042638f};
__constant__ float GAMMA_TANH[10] = {
    0.715951561820333f, 1.039358092507381f, 0.948607106485449f, 0.715951555650637f,
    0.715951343627955f, 1.039354251532628f, 0.484068718800797f, 0.715951452277779f,
    1.039355768740833f, 1.446433070133343f};

// -------- generic C[M,N] = act(A[M,K] @ B[N,K]^T + bias), f32 WMMA ----------
__global__ __launch_bounds__(32) void gemm_bt_wmma(
    const float* __restrict__ A, const float* __restrict__ Bm,
    const float* __restrict__ bias, float* __restrict__ C,
    int M, int N, int K, int act)
{
    const int lane = threadIdx.x;
    const int half = lane >> 4, l16 = lane & 15;
    const int n0 = blockIdx.x * 16, m0 = blockIdx.y * 16;

    const float* __restrict__ arow = A  + (size_t)(m0 + l16) * K;
    const float* __restrict__ brow = Bm + (size_t)(n0 + l16) * K;

    v8f acc = {};
    for (int k = 0; k < K; k += 4) {
        const int kk = k + 2 * half;
        v2f a, b;
        a.x = arow[kk]; a.y = arow[kk + 1];
        b.x = brow[kk]; b.y = brow[kk + 1];
        acc = __builtin_amdgcn_wmma_f32_16x16x4_f32(
            false, a, false, b, (short)0, acc, false, false);
    }
    const int col = n0 + l16;
    const float bv = bias ? bias[col] : 0.0f;
#pragma unroll
    for (int r = 0; r < 8; ++r) {
        const int row = m0 + r + 8 * half;
        float v = acc[r] + bv;
        if (act == 1) v = tanhf(v);
        C[(size_t)row * N + col] = v;
    }
}

// -------- z_var = exp(z_logvar) ---------------------------------------------
__global__ void exp_k(const float* __restrict__ in, float* __restrict__ out, int n)
{
    int i = blockIdx.x * blockDim.x + threadIdx.x;
    if (i < n) out[i] = expf(in[i]);
}

// -------- diag(cov3)[b,i] = sum_k w3[i,k]^2 * z_var[b,k], clipped -----------
__global__ void diag3_k(const float* __restrict__ w3, const float* __restrict__ zvar,
                        float* __restrict__ dg)
{
    int i = blockIdx.x * blockDim.x + threadIdx.x;   // over 64*512
    if (i >= 64 * 512) return;
    const int b = i >> 9, r = i & 511;
    const float* __restrict__ w = w3 + (size_t)r * 64;
    const float* __restrict__ v = zvar + (size_t)b * 64;
    float s = 0.0f;
#pragma unroll 8
    for (int k = 0; k < 64; ++k) s += w[k] * w[k] * v[k];
    dg[i] = fmaxf(s, EPS);
}

// -------- tanh moment coefficients A0..A3 over 10 gammas --------------------
__global__ void tanh_mom_k(const float* __restrict__ m3, const float* __restrict__ dvar,
                           float* __restrict__ A0, float* __restrict__ A1,
                           float* __restrict__ A2, float* __restrict__ A3, int n)
{
    int i = blockIdx.x * blockDim.x + threadIdx.x;
    if (i >= n) return;
    const float m = m3[i], var = dvar[i];
    float a0 = 0.f, a1 = 0.f, a2 = 0.f, a3 = 0.f;
#pragma unroll
    for (int g = 0; g < 10; ++g) {
        const float gam = GAMMA_TANH[g];
        const float ivh = var + 0.5f / (gam * gam);
        const float s = sqrtf(ivh);
        const float mu = m / s;
        const float Bv = expf(-0.5f * mu * mu) * INV_SQRT_2PI / s;
        const float Cv = 0.5f * (1.0f + erff(mu * 0.70710678118654752f));
        a0 += 2.0f * Cv - 1.0f;
        a1 += 2.0f * Bv;
        a2 += -Bv * m / ivh;
        a3 += (1.0f / 3.0f) * Bv * (m * m - ivh) / (ivh * ivh);
    }
    A0[i] = a0 * 0.1f; A1[i] = a1 * 0.1f; A2[i] = a2 * 0.1f; A3[i] = a3 * 0.1f;
}

// -------- fused cov_h3[b] tile: WMMA cov3 + moment epilogue -----------------
__global__ __launch_bounds__(32) void covh3_k(
    const float* __restrict__ w3,   // [512,64]
    const float* __restrict__ zvar, // [64,64]
    const float* __restrict__ dvar, // [64,512] clipped diag(cov3)
    const float* __restrict__ A1, const float* __restrict__ A2,
    const float* __restrict__ A3,
    float* __restrict__ cov)        // [64,512,512]
{
    const int lane = threadIdx.x;
    const int half = lane >> 4, l16 = lane & 15;
    const int j0 = blockIdx.x * 16, i0 = blockIdx.y * 16, b = blockIdx.z;

    const float* __restrict__ zv = zvar + (size_t)b * 64;
    const float* __restrict__ wi = w3 + (size_t)(i0 + l16) * 64;
    const float* __restrict__ wj = w3 + (size_t)(j0 + l16) * 64;

    v8f acc = {};
#pragma unroll
    for (int k = 0; k < 64; k += 4) {
        const int kk = k + 2 * half;
        v2f a, bf;
        a.x = wi[kk] * zv[kk]; a.y = wi[kk + 1] * zv[kk + 1];
        bf.x = wj[kk];         bf.y = wj[kk + 1];
        acc = __builtin_amdgcn_wmma_f32_16x16x4_f32(
            false, a, false, bf, (short)0, acc, false, false);
    }

    const int j = j0 + l16;
    const size_t bo = (size_t)b * 512;
    const float a1j = A1[bo + j], a2j = A2[bo + j], a3j = A3[bo + j], dj = dvar[bo + j];
    float* __restrict__ covb = cov + (size_t)b * 512 * 512;
#pragma unroll
    for (int r = 0; r < 8; ++r) {
        const int i = i0 + r + 8 * half;
        float c = acc[r];
        if (i == j) c = fmaxf(c, EPS);              // clipped diagonal
        const float a1i = A1[bo + i], a2i = A2[bo + i], a3i = A3[bo + i], di = dvar[bo + i];
        const float c2 = c * c;
        const float post = a1i * a1j * c
                         + a2i * a2j * (2.0f * c2)
                         + a3i * a3j * (6.0f * c2 * c + 9.0f * di * dj * c);
        covb[(size_t)i * 512 + j] = post;
    }
}

// -------- diag(cov41)[b,o] = w41[o,:] @ cov_h3[b] @ w41[o,:] ----------------
// cov_h3 is symmetric, so compute G^T[j,o] = sum_i cov[b,j,i] * w41[o,i]:
// cov rows become the (row-major contiguous) WMMA A operand, w41 o-tile
// (staged once in LDS) is the B operand. 4-way j-tile register blocking
// amortizes the B fragment over 4 WMMAs. Accumulator column index IS the
// output 'o', so the final weighted reduction is one scalar per lane plus
// a single shfl_xor(16).
#define LSTR 516   // LDS row stride (floats): 516 % 64 == 4 -> conflict-free
__global__ __launch_bounds__(128) void diag41_k(
    const float* __restrict__ w41,  // [784,512]
    const float* __restrict__ cov,  // [64,512,512]
    float* __restrict__ dg)         // [64,784]
{
    __shared__ float Atile[16 * LSTR];
    const int tid  = threadIdx.x;
    const int wave = tid >> 5;              // 4 waves, one batch each
    const int lane = tid & 31;
    const int half = lane >> 4, l16 = lane & 15;
    const int o0 = blockIdx.x * 16;
    const int b  = blockIdx.y * 4 + wave;

    // stage the 16x512 w41 o-tile in LDS (shared by all 4 waves)
    for (int t = tid; t < 16 * 512; t += 128) {
        const int row = t >> 9, cx = t & 511;
        Atile[row * LSTR + cx] = w41[(size_t)(o0 + row) * 512 + cx];
    }
    __syncthreads();

    const float* __restrict__ covb = cov + (size_t)b * 512 * 512;
    const int aoff = l16 * LSTR;            // this lane's w41 row in LDS
    float part = 0.0f;                      // partial for o = o0 + l16

    for (int jblk = 0; jblk < 512; jblk += 64) {
        // 4 cov-row streams (A operand), one per j-tile in this block
        const float* __restrict__ c0 = covb + (size_t)(jblk +  0 + l16) * 512;
        const float* __restrict__ c1 = covb + (size_t)(jblk + 16 + l16) * 512;
        const float* __restrict__ c2 = covb + (size_t)(jblk + 32 + l16) * 512;
        const float* __restrict__ c3 = covb + (size_t)(jblk + 48 + l16) * 512;

        v8f acc0 = {}, acc1 = {}, acc2 = {}, acc3 = {};
        for (int k = 0; k < 512; k += 4) {
            const int kk = k + 2 * half;
            v2f bf;                          // B = w41 o-tile from LDS
            bf.x = Atile[aoff + kk];
            bf.y = Atile[aoff + kk + 1];
            v2f a0, a1, a2, a3;              // A = cov rows, contiguous b64
            a0.x = c0[kk]; a0.y = c0[kk + 1];
            a1.x = c1[kk]; a1.y = c1[kk + 1];
            a2.x = c2[kk]; a2.y = c2[kk + 1];
            a3.x = c3[kk]; a3.y = c3[kk + 1];
            acc0 = __builtin_amdgcn_wmma_f32_16x16x4_f32(false, a0, false, bf, (short)0, acc0, false, false);
            acc1 = __builtin_amdgcn_wmma_f32_16x16x4_f32(false, a1, false, bf, (short)0, acc1, false, false);
            acc2 = __builtin_amdgcn_wmma_f32_16x16x4_f32(false, a2, false, bf, (short)0, acc2, false, false);
            acc3 = __builtin_amdgcn_wmma_f32_16x16x4_f32(false, a3, false, bf, (short)0, acc3, false, false);
        }
        // weighted reduction: part += G^T[j, o] * w41[o, j]
#pragma unroll
        for (int r = 0; r < 8; ++r) {
            const int jr = r + 8 * half;
            part += acc0[r] * Atile[aoff + jblk +  0 + jr];
            part += acc1[r] * Atile[aoff + jblk + 16 + jr];
            part += acc2[r] * Atile[aoff + jblk + 32 + jr];
            part += acc3[r] * Atile[aoff + jblk + 48 + jr];
        }
    }
    // combine the two half-wave row groups
    part += __shfl_xor(part, 16, 32);
    if (half == 0)
        dg[(size_t)b * 784 + o0 + l16] = part;
}

// -------- probs = sigmoid moment A0 over 10 gammas --------------------------
__global__ void probs_k(const float* __restrict__ m41, const float* __restrict__ dvar,
                        float* __restrict__ out, int n)
{
    int i = blockIdx.x * blockDim.x + threadIdx.x;
    if (i >= n) return;
    const float m = m41[i];
    const float var = fmaxf(dvar[i], EPS);
    float a0 = 0.0f;
#pragma unroll
    for (int g = 0; g < 10; ++g) {
        const float gam = GAMMA_SIGMOID[g];
        const float ivh = var + 0.5f / (gam * gam);
        const float mu = m / sqrtf(ivh);
        a0 += 0.5f * (1.0f + erff(mu * 0.70710678118654752f));
    }
    out[i] = a0 * 0.1f;
}

extern "C" void kernel_launch(void* const* d_in, const int* in_sizes, int n_in,
                              void* d_out, int out_size, void* d_ws, size_t ws_size,
                              hipStream_t stream)
{
    const float* x   = (const float*)d_in[0];   // [64,784]
    const float* w1  = (const float*)d_in[1];   // [512,784]
    const float* b1  = (const float*)d_in[2];
    const float* w21 = (const float*)d_in[3];   // [64,512]
    const float* b21 = (const float*)d_in[4];
    const float* w22 = (const float*)d_in[5];   // [64,512]
    const float* b22 = (const float*)d_in[6];
    const float* w3  = (const float*)d_in[7];   // [512,64]
    const float* b3  = (const float*)d_in[8];
    const float* w41 = (const float*)d_in[9];   // [784,512]
    const float* b41 = (const float*)d_in[10];

    float* out      = (float*)d_out;
    float* o_zmean  = out;                      // 64*64
    float* o_zlogv  = out + 4096;               // 64*64
    float* o_probs  = out + 8192;               // 64*784

    float* W    = (float*)d_ws;
    float* h1   = W;                 // 32768
    float* zvar = h1 + 32768;        // 4096
    float* m3   = zvar + 4096;       // 32768
    float* dg3  = m3 + 32768;        // 32768
    float* A0   = dg3 + 32768;       // 32768
    float* A1   = A0 + 32768;        // 32768
    float* A2   = A1 + 32768;        // 32768
    float* A3   = A2 + 32768;        // 32768
    float* m41  = A3 + 32768;        // 50176
    float* dg41 = m41 + 50176;       // 50176
    float* cov  = dg41 + 50176;      // 64*512*512 = 16777216

    // h1 = tanh(x @ w1^T + b1)
    gemm_bt_wmma<<<dim3(512 / 16, 64 / 16), 32, 0, stream>>>(x, w1, b1, h1, 64, 512, 784, 1);
    // z_mean / z_logvar
    gemm_bt_wmma<<<dim3(64 / 16, 64 / 16), 32, 0, stream>>>(h1, w21, b21, o_zmean, 64, 64, 512, 0);
    gemm_bt_wmma<<<dim3(64 / 16, 64 / 16), 32, 0, stream>>>(h1, w22, b22, o_zlogv, 64, 64, 512, 0);
    exp_k<<<(4096 + 255) / 256, 256, 0, stream>>>(o_zlogv, zvar, 4096);
    // m3 = z_mean @ w3^T + b3
    gemm_bt_wmma<<<dim3(512 / 16, 64 / 16), 32, 0, stream>>>(o_zmean, w3, b3, m3, 64, 512, 64, 0);
    // clipped diag(cov3)
    diag3_k<<<(64 * 512 + 255) / 256, 256, 0, stream>>>(w3, zvar, dg3);
    // tanh moments: A0 = m_h3
    tanh_mom_k<<<(32768 + 255) / 256, 256, 0, stream>>>(m3, dg3, A0, A1, A2, A3, 32768);
    // m41 = m_h3 @ w41^T + b41
    gemm_bt_wmma<<<dim3(784 / 16, 64 / 16), 32, 0, stream>>>(A0, w41, b41, m41, 64, 784, 512, 0);
    // cov_h3 (fused cov3 WMMA + moment epilogue)
    covh3_k<<<dim3(32, 32, 64), 32, 0, stream>>>(w3, zvar, dg3, A1, A2, A3, cov);
    // diag(cov41) via batched WMMA GEMM (cov symmetric -> cov as A operand)
    diag41_k<<<dim3(784 / 16, 64 / 4), 128, 0, stream>>>(w41, cov, dg41);
    // probs
    probs_k<<<(50176 + 255) / 256, 256, 0, stream>>>(m41, dg41, o_probs, 50176);
}